// InflectionModule_1168231105128
// MI455X (gfx1250) — compile-verified
//
#include <hip/hip_runtime.h>
#include <math.h>

// ---- model dims ----
#define HS   100        // STATE_SIZE
#define ATTD 100        // ATTENTION_SIZE
#define ED   32         // EMBEDDINGS_SIZE
#define VV   50257      // char vocab
#define NN   2048       // source length
#define TD   2048       // target length
#define TTG  64         // number of tags
#define DXIN 332        // 3H + E decoder input width
#define NWAVE 8         // waves per loss block

typedef __attribute__((ext_vector_type(2))) float v2f;
typedef __attribute__((ext_vector_type(8))) float v8f;

// =====================================================================
// Kernel 1: the three encoder LSTMs (fwd char / bwd char / tag), one
// block each. Sequential over time; 400 gate dot-products in parallel.
// =====================================================================
__global__ __launch_bounds__(512) void encoder_kernel(
    const int* src_ids, const int* tag_ids,
    const float* emb_char, const float* emb_tag,
    const float* fWih, const float* fWhh, const float* fb,
    const float* bWih, const float* bWhh, const float* bb,
    const float* tWih, const float* tWhh, const float* tb,
    float* enc, float* tv) {
  __shared__ float x[ED], h[HS], c[HS], z[4 * HS];
  const int tid = threadIdx.x;
  const int blk = blockIdx.x;

  const float *Wih, *Whh, *bias, *emb;
  const int* ids;
  int steps;
  if (blk == 0)      { Wih = fWih; Whh = fWhh; bias = fb; ids = src_ids; emb = emb_char; steps = NN; }
  else if (blk == 1) { Wih = bWih; Whh = bWhh; bias = bb; ids = src_ids; emb = emb_char; steps = NN; }
  else               { Wih = tWih; Whh = tWhh; bias = tb; ids = tag_ids; emb = emb_tag;  steps = TTG; }

  if (tid < HS) { h[tid] = 0.f; c[tid] = 0.f; }
  __syncthreads();

  for (int t = 0; t < steps; ++t) {
    const int pos = (blk == 1) ? (steps - 1 - t) : t;   // bwd scans reversed
    if (tid < ED) x[tid] = emb[ids[pos] * ED + tid];
    __syncthreads();
    if (tid < 4 * HS) {
      float s = bias[tid];
      const float* w = Wih + tid * ED;
      #pragma unroll 8
      for (int k = 0; k < ED; ++k) s += x[k] * w[k];
      const float* wh = Whh + tid * HS;
      for (int k = 0; k < HS; ++k) s += h[k] * wh[k];
      z[tid] = s;
    }
    __syncthreads();
    if (tid < HS) {
      const float ig = 1.f / (1.f + __expf(-z[tid]));
      const float fg = 1.f / (1.f + __expf(-z[HS + tid]));
      const float gg = tanhf(z[2 * HS + tid]);
      const float og = 1.f / (1.f + __expf(-z[3 * HS + tid]));
      const float cn = fg * c[tid] + ig * gg;
      c[tid] = cn;
      const float hn = og * tanhf(cn);
      h[tid] = hn;
      if (blk == 0)      enc[pos * (2 * HS) + tid] = hn;        // yf -> cols [0,100)
      else if (blk == 1) enc[pos * (2 * HS) + HS + tid] = hn;   // yb -> cols [100,200)
      else               tv[t * HS + tid] = hn;
    }
    __syncthreads();
  }
}

// =====================================================================
// Kernel 2: tag self-attention residual + tag_w1dt precompute (tiny).
// =====================================================================
__global__ __launch_bounds__(256) void tag_attn_kernel(
    const float* tv, const float* tag_attn_w1, float* tagv, float* tag_w1dt) {
  __shared__ float sa[TTG][TTG];
  const int tid = threadIdx.x;
  for (int idx = tid; idx < TTG * TTG; idx += 256) {
    const int s2 = idx / TTG, t2 = idx % TTG;
    const float* a = tv + s2 * HS;
    const float* b = tv + t2 * HS;
    float s = 0.f;
    for (int k = 0; k < HS; ++k) s += a[k] * b[k];
    sa[s2][t2] = s;
  }
  __syncthreads();
  if (tid < TTG) {   // softmax along t2 for each row s
    float m = sa[tid][0];
    for (int j = 1; j < TTG; ++j) m = fmaxf(m, sa[tid][j]);
    float su = 0.f;
    for (int j = 0; j < TTG; ++j) { const float e = __expf(sa[tid][j] - m); sa[tid][j] = e; su += e; }
    const float inv = 1.f / su;
    for (int j = 0; j < TTG; ++j) sa[tid][j] *= inv;
  }
  __syncthreads();
  // tagv[t][h] = tv[t][h] + sum_s sa[s][t] * tv[s][h]
  for (int idx = tid; idx < TTG * HS; idx += 256) {
    const int t2 = idx / HS, hh = idx % HS;
    float s = tv[t2 * HS + hh];
    for (int s2 = 0; s2 < TTG; ++s2) s += sa[s2][t2] * tv[s2 * HS + hh];
    tagv[idx] = s;
  }
  __syncthreads();
  // tag_w1dt = tagv @ tag_attn_w1.T
  for (int idx = tid; idx < TTG * ATTD; idx += 256) {
    const int s2 = idx / ATTD, a = idx % ATTD;
    const float* tg = tagv + s2 * HS;
    const float* w = tag_attn_w1 + a * HS;
    float s = 0.f;
    for (int k = 0; k < HS; ++k) s += tg[k] * w[k];
    tag_w1dt[idx] = s;
  }
}

// =====================================================================
// Kernel 3: w1dt = enc @ attn_w1.T   (2048x200 @ 200x100, 41M MAC)
// =====================================================================
__global__ void w1dt_kernel(const float* enc, const float* attn_w1, float* w1dt) {
  const int idx = blockIdx.x * blockDim.x + threadIdx.x;
  if (idx >= NN * ATTD) return;
  const int row = idx / ATTD, col = idx % ATTD;
  const float* e = enc + row * (2 * HS);
  const float* w = attn_w1 + col * (2 * HS);
  float s = 0.f;
  for (int k = 0; k < 2 * HS; ++k) s += e[k] * w[k];
  w1dt[idx] = s;
}

// =====================================================================
// Kernel 4: decoder recurrence (sequential in t). One block, 1024 thr.
// Stores h_t into Hs[t] for the parallel WMMA loss pass.
// =====================================================================
__global__ __launch_bounds__(1024) void decoder_kernel(
    const float* enc, const float* tagv, const float* tag_w1dt, const float* w1dt,
    const float* emb_char, const int* out_ids,
    const float* dec_Wih, const float* dec_Whh, const float* dec_b,
    const float* attn_w2, const float* attn_w3, const float* attn_v,
    const float* tag_attn_w2, const float* tag_attn_v,
    float* Hs) {
  __shared__ float h[HS], c[HS];
  __shared__ float q[ATTD], tsc[TTG], tctx[HS], nstate[2 * HS], q2[ATTD], q3[ATTD];
  __shared__ float scores[NN];
  __shared__ float redv[1024];
  __shared__ int   redi[1024];
  __shared__ float ctx[2 * HS], xin[DXIN], z[4 * HS];
  __shared__ float prevatt[5], newprev[5];
  const int tid = threadIdx.x;

  // ---- initial decoder step: x0 = [enc[-1], tagv[-1], emb_char[0]], h=c=0
  if (tid < 2 * HS)                    xin[tid] = enc[(NN - 1) * (2 * HS) + tid];
  if (tid >= 2 * HS && tid < 3 * HS)   xin[tid] = tagv[(TTG - 1) * HS + (tid - 2 * HS)];
  if (tid >= 3 * HS && tid < DXIN)     xin[tid] = emb_char[tid - 3 * HS];
  if (tid < HS) { h[tid] = 0.f; c[tid] = 0.f; }
  if (tid < 5) prevatt[tid] = 0.f;
  __syncthreads();
  if (tid < 4 * HS) {
    float s = dec_b[tid];
    const float* w = dec_Wih + tid * DXIN;
    for (int k = 0; k < DXIN; ++k) s += xin[k] * w[k];
    z[tid] = s;            // h == 0 -> Whh term vanishes
  }
  __syncthreads();
  if (tid < HS) {
    const float ig = 1.f / (1.f + __expf(-z[tid]));
    const float fg = 1.f / (1.f + __expf(-z[HS + tid]));
    const float gg = tanhf(z[2 * HS + tid]);
    const float og = 1.f / (1.f + __expf(-z[3 * HS + tid]));
    const float cn = fg * c[tid] + ig * gg;
    c[tid] = cn;
    h[tid] = og * tanhf(cn);
  }
  __syncthreads();

  for (int t = 0; t < TD; ++t) {
    // -- tag attention query: q = [h,c] @ tag_attn_w2.T
    if (tid < ATTD) {
      const float* w = tag_attn_w2 + tid * (2 * HS);
      float s = 0.f;
      for (int k = 0; k < HS; ++k) s += h[k] * w[k];
      for (int k = 0; k < HS; ++k) s += c[k] * w[HS + k];
      q[tid] = s;
    }
    __syncthreads();
    if (tid < TTG) {
      const float* tw = tag_w1dt + tid * ATTD;
      float s = 0.f;
      for (int a = 0; a < ATTD; ++a) s += tanhf(tw[a] + q[a]) * tag_attn_v[a];
      tsc[tid] = s;
    }
    __syncthreads();
    if (tid == 0) {        // softmax over 64 tag scores (tiny, serial)
      float m = tsc[0];
      for (int s2 = 1; s2 < TTG; ++s2) m = fmaxf(m, tsc[s2]);
      float su = 0.f;
      for (int s2 = 0; s2 < TTG; ++s2) { const float e = __expf(tsc[s2] - m); tsc[s2] = e; su += e; }
      const float inv = 1.f / su;
      for (int s2 = 0; s2 < TTG; ++s2) tsc[s2] *= inv;
    }
    __syncthreads();
    if (tid < HS) {
      float s = 0.f;
      for (int s2 = 0; s2 < TTG; ++s2) s += tsc[s2] * tagv[s2 * HS + tid];
      tctx[tid] = s;
    }
    __syncthreads();
    if (tid < 2 * HS) nstate[tid] = ((tid < HS) ? h[tid] : c[tid - HS]) + tctx[tid % HS];
    __syncthreads();
    // -- char attention queries
    if (tid < ATTD) {
      const float* w = attn_w2 + tid * (2 * HS);
      float s = 0.f;
      for (int k = 0; k < 2 * HS; ++k) s += nstate[k] * w[k];
      q2[tid] = s;
      const float* w3 = attn_w3 + tid * 5;
      float s3 = 0.f;
      for (int k = 0; k < 5; ++k) s3 += prevatt[k] * w3[k];
      q3[tid] = s3;
    }
    __syncthreads();
    // -- scores over N = 2048 (2 rows per thread)
    for (int n = tid; n < NN; n += 1024) {
      const float* wd = w1dt + n * ATTD;
      float s = 0.f;
      for (int a = 0; a < ATTD; ++a) s += tanhf(wd[a] + q2[a] + q3[a]) * attn_v[a];
      scores[n] = s;
    }
    __syncthreads();
    // -- max + first-argmax reduction
    {
      const float v0 = scores[tid], v1 = scores[tid + 1024];
      float mv; int mi;
      if (v1 > v0) { mv = v1; mi = tid + 1024; } else { mv = v0; mi = tid; }
      redv[tid] = mv; redi[tid] = mi;
      __syncthreads();
      for (int off = 512; off > 0; off >>= 1) {
        if (tid < off) {
          const float ov = redv[tid + off]; const int oi = redi[tid + off];
          if (ov > redv[tid] || (ov == redv[tid] && oi < redi[tid])) { redv[tid] = ov; redi[tid] = oi; }
        }
        __syncthreads();
      }
    }
    const float gmax = redv[0];
    const int   gidx = redi[0];
    __syncthreads();
    // -- exp + sum
    {
      const float e0 = __expf(scores[tid] - gmax);
      const float e1 = __expf(scores[tid + 1024] - gmax);
      scores[tid] = e0; scores[tid + 1024] = e1;
      redv[tid] = e0 + e1;
      __syncthreads();
      for (int off = 512; off > 0; off >>= 1) {
        if (tid < off) redv[tid] += redv[tid + off];
        __syncthreads();
      }
    }
    const float inv = 1.f / redv[0];
    __syncthreads();
    scores[tid] *= inv; scores[tid + 1024] *= inv;
    __syncthreads();
    // -- 5-wide previous-attention window
    if (tid < 5) {
      int st = gidx - 1;
      if (st < 0) st = 0;
      if (st > NN - 6) st = NN - 6;
      newprev[tid] = scores[st + tid];
    }
    // -- ctx = aw @ enc   (200 cols x 5-way split over n)
    if (tid < 1000) {
      const int g = tid / 5, l = tid % 5;
      float s = 0.f;
      for (int n = l; n < NN; n += 5) s += scores[n] * enc[n * (2 * HS) + g];
      redv[tid] = s;
    }
    __syncthreads();
    if (tid < 2 * HS)
      ctx[tid] = redv[tid * 5] + redv[tid * 5 + 1] + redv[tid * 5 + 2] + redv[tid * 5 + 3] + redv[tid * 5 + 4];
    __syncthreads();
    // -- LSTM input [ctx, tag_ctx, prev_emb]
    if (tid < 2 * HS)                    xin[tid] = ctx[tid];
    if (tid >= 2 * HS && tid < 3 * HS)   xin[tid] = tctx[tid - 2 * HS];
    if (tid >= 3 * HS && tid < DXIN) {
      const int pid = (t == 0) ? 0 : out_ids[t - 1];
      xin[tid] = emb_char[pid * ED + (tid - 3 * HS)];
    }
    if (tid < 5) prevatt[tid] = newprev[tid];
    __syncthreads();
    if (tid < 4 * HS) {
      float s = dec_b[tid];
      const float* w = dec_Wih + tid * DXIN;
      for (int k = 0; k < DXIN; ++k) s += xin[k] * w[k];
      const float* wh = dec_Whh + tid * HS;
      for (int k = 0; k < HS; ++k) s += h[k] * wh[k];
      z[tid] = s;
    }
    __syncthreads();
    if (tid < HS) {
      const float ig = 1.f / (1.f + __expf(-z[tid]));
      const float fg = 1.f / (1.f + __expf(-z[HS + tid]));
      const float gg = tanhf(z[2 * HS + tid]);
      const float og = 1.f / (1.f + __expf(-z[3 * HS + tid]));
      const float cn = fg * c[tid] + ig * gg;
      c[tid] = cn;
      const float hn = og * tanhf(cn);
      h[tid] = hn;
      Hs[t * HS + tid] = hn;       // feed the parallel WMMA loss pass
    }
    __syncthreads();
  }
}

// =====================================================================
// Kernel 5: zero the output accumulator.
// =====================================================================
__global__ void zero_kernel(float* out, int n) {
  const int i = blockIdx.x * blockDim.x + threadIdx.x;
  if (i < n) out[i] = 0.f;
}

// =====================================================================
// Kernel 6: fused  logits = Hs @ out_W.T + out_b  -> online log-softmax
// -> per-row NLL, via V_WMMA_F32_16X16X4_F32.
//   * 8 waves per block; wave w owns vocab tiles w, w+8, ... (1024
//     independent waves streaming the L2-resident 20 MB out_W).
//   * B operands batch-loaded (25 b64 loads in flight, one wait).
//   * TWO independent WMMA accumulator chains (even/odd k-steps) to
//     break the 25-deep C-operand RAW chain; merged with 8 f32 adds.
//   * Per-wave online (max, sumexp, logit[oid]) stats, merged per row
//     across waves at the end; single atomicAdd per row.
// =====================================================================
__global__ __launch_bounds__(32 * NWAVE) void loss_kernel(
    const float* __restrict__ Hs, const float* __restrict__ out_W,
    const float* __restrict__ out_b, const int* __restrict__ out_ids,
    const float* __restrict__ weight, float* out) {
  __shared__ float tile[NWAVE][16][17];
  __shared__ float mmax[NWAVE][16], msum[NWAVE][16], mlog[NWAVE][16];
  const int tid  = threadIdx.x;
  const int wv   = tid >> 5;               // wave id 0..7
  const int lane = tid & 31;               // wave32 lane
  const int m0   = blockIdx.x * 16;        // 128 blocks x 16 rows = 2048
  const int mloc = lane & 15;
  const int koff = (lane >> 4) * 2;        // A/B K split across lane halves

  // Preload the whole 16x100 A block (reused for ~393 vocab tiles/wave).
  // A layout (16x16x4 f32): lane l holds A[l%16][k0 + 2*(l/16) + v]
  v2f aReg[25];
  #pragma unroll
  for (int kk = 0; kk < 25; ++kk) {
    const float* p = Hs + (size_t)(m0 + mloc) * HS + kk * 4 + koff;
    aReg[kk].x = p[0];
    aReg[kk].y = p[1];
  }

  float rowmax = -1e30f, rowsum = 0.f, rowlogit = 0.f;
  const int oid = out_ids[m0 + mloc];      // target id for row (lanes<16 use it)
  const int ntiles = (VV + 15) / 16;       // 3142

  for (int tI = wv; tI < ntiles; tI += NWAVE) {
    const int n0 = tI * 16;
    const int n  = n0 + mloc;
    const bool nval = (n < VV);
    const float* bp = out_W + (size_t)n * HS + koff;

    // ---- batch-load B tile (+ bias): independent loads, one wait ----
    // B layout (4x16): lane l holds B[2*(l/16)+v][l%16] = out_W[n][k]
    const float ob = nval ? out_b[n] : 0.f;
    v2f bReg[25];
    #pragma unroll
    for (int kk = 0; kk < 25; ++kk) {
      if (nval) { bReg[kk].x = bp[kk * 4]; bReg[kk].y = bp[kk * 4 + 1]; }
      else      { bReg[kk].x = 0.f;        bReg[kk].y = 0.f;            }
    }
    if (nval && tI + NWAVE < ntiles)
      __builtin_prefetch(bp + NWAVE * 16 * HS, 0, 3);   // next stripe -> global_prefetch

    // ---- two independent 12/13-deep WMMA chains (break C RAW chain) ----
    v8f acc0 = {}, acc1 = {};
    #pragma unroll
    for (int kk = 0; kk < 24; kk += 2) {
      acc0 = __builtin_amdgcn_wmma_f32_16x16x4_f32(
          false, aReg[kk],     false, bReg[kk],     (short)0, acc0, false, false);
      acc1 = __builtin_amdgcn_wmma_f32_16x16x4_f32(
          false, aReg[kk + 1], false, bReg[kk + 1], (short)0, acc1, false, false);
    }
    acc0 = __builtin_amdgcn_wmma_f32_16x16x4_f32(
        false, aReg[24], false, bReg[24], (short)0, acc0, false, false);

    // D layout: VGPR r holds rows (r + 8*(lane/16)), col lane%16
    const int mbase = (lane >> 4) * 8;
    #pragma unroll
    for (int r = 0; r < 8; ++r)
      tile[wv][mbase + r][mloc] = acc0[r] + acc1[r] + ob;
    // wave-private LDS tile: RAW within this wave only, no block barrier
    asm volatile("s_wait_dscnt 0x0" ::: "memory");

    if (lane < 16) {   // online softmax update, one lane per row
      float tmax = -1e30f;
      for (int j = 0; j < 16; ++j)
        if (n0 + j < VV) { const float x = tile[wv][lane][j]; if (x > tmax) tmax = x; }
      const float nm = (rowmax > tmax) ? rowmax : tmax;
      float s = rowsum * __expf(rowmax - nm);
      for (int j = 0; j < 16; ++j)
        if (n0 + j < VV) s += __expf(tile[wv][lane][j] - nm);
      rowsum = s; rowmax = nm;
      if (oid >= n0 && oid < n0 + 16) rowlogit = tile[wv][lane][oid - n0];
    }
  }

  // ---- merge the 8 per-wave online stats for each of the 16 rows ----
  if (lane < 16) {
    mmax[wv][lane] = rowmax;
    msum[wv][lane] = rowsum;
    mlog[wv][lane] = rowlogit;   // exactly one wave captured oid's tile
  }
  __syncthreads();
  if (tid < 16) {
    float M = -1e30f;
    for (int w = 0; w < NWAVE; ++w) M = fmaxf(M, mmax[w][tid]);
    float S = 0.f, L = 0.f;
    for (int w = 0; w < NWAVE; ++w) {
      S += msum[w][tid] * __expf(mmax[w][tid] - M);
      L += mlog[w][tid];
    }
    const float loss = M + __logf(S) - L;   // -log_softmax[oid]
    atomicAdd(out, loss * weight[0]);
  }
}

// =====================================================================
extern "C" void kernel_launch(void* const* d_in, const int* in_sizes, int n_in,
                              void* d_out, int out_size, void* d_ws, size_t ws_size,
                              hipStream_t stream) {
  const int*   src_ids     = (const int*)d_in[0];
  const int*   tag_ids     = (const int*)d_in[1];
  const int*   out_ids     = (const int*)d_in[2];
  const float* weight      = (const float*)d_in[3];
  const float* emb_char    = (const float*)d_in[4];
  const float* emb_tag     = (const float*)d_in[5];
  const float* enc_f_Wih   = (const float*)d_in[6];
  const float* enc_f_Whh   = (const float*)d_in[7];
  const float* enc_f_b     = (const float*)d_in[8];
  const float* enc_b_Wih   = (const float*)d_in[9];
  const float* enc_b_Whh   = (const float*)d_in[10];
  const float* enc_b_b     = (const float*)d_in[11];
  const float* tag_Wih     = (const float*)d_in[12];
  const float* tag_Whh     = (const float*)d_in[13];
  const float* tag_b       = (const float*)d_in[14];
  const float* dec_Wih     = (const float*)d_in[15];
  const float* dec_Whh     = (const float*)d_in[16];
  const float* dec_b       = (const float*)d_in[17];
  const float* attn_w1     = (const float*)d_in[18];
  const float* attn_w2     = (const float*)d_in[19];
  const float* attn_w3     = (const float*)d_in[20];
  const float* attn_v      = (const float*)d_in[21];
  const float* tag_attn_w1 = (const float*)d_in[22];
  const float* tag_attn_w2 = (const float*)d_in[23];
  const float* tag_attn_v  = (const float*)d_in[24];
  const float* out_W       = (const float*)d_in[25];
  const float* out_b       = (const float*)d_in[26];

  // workspace layout (floats): ~3.35 MB total
  float* ws       = (float*)d_ws;
  float* enc      = ws;                       // N x 2H      = 409600
  float* tv       = enc + NN * 2 * HS;        // TT x H      =   6400
  float* tagv     = tv + TTG * HS;            // TT x H      =   6400
  float* tag_w1dt = tagv + TTG * HS;          // TT x ATT    =   6400
  float* w1dt     = tag_w1dt + TTG * ATTD;    // N x ATT     = 204800
  float* Hsbuf    = w1dt + NN * ATTD;         // T x H       = 204800

  encoder_kernel<<<3, 512, 0, stream>>>(
      src_ids, tag_ids, emb_char, emb_tag,
      enc_f_Wih, enc_f_Whh, enc_f_b,
      enc_b_Wih, enc_b_Whh, enc_b_b,
      tag_Wih, tag_Whh, tag_b, enc, tv);

  tag_attn_kernel<<<1, 256, 0, stream>>>(tv, tag_attn_w1, tagv, tag_w1dt);

  w1dt_kernel<<<(NN * ATTD + 255) / 256, 256, 0, stream>>>(enc, attn_w1, w1dt);

  decoder_kernel<<<1, 1024, 0, stream>>>(
      enc, tagv, tag_w1dt, w1dt, emb_char, out_ids,
      dec_Wih, dec_Whh, dec_b,
      attn_w2, attn_w3, attn_v,
      tag_attn_w2, tag_attn_v, Hsbuf);

  zero_kernel<<<1, 64, 0, stream>>>((float*)d_out, out_size);

  loss_kernel<<<TD / 16, 32 * NWAVE, 0, stream>>>(
      Hsbuf, out_W, out_b, out_ids, weight, (float*)d_out);
}